// CM_Module_39934605919003
// MI455X (gfx1250) — compile-verified
//
#include <hip/hip_runtime.h>
#include <math.h>

typedef float v2f __attribute__((ext_vector_type(2)));
typedef float v8f __attribute__((ext_vector_type(8)));

#define BB 8
#define CC 128
#define FF 8
#define FR 7          // f - 1
#define HW 4096       // 64*64
#define HIN 256       // input mask H=W
#define NPART 512     // gs partial slots per (b,f): 2 c-halves * 256 pixel tiles

// workspace layout (float offsets)
#define WS_VTR   0                          // [8][4096]
#define WS_VR    (WS_VTR + BB*HW)           // [8][7][4096]
#define WS_CM    (WS_VR  + BB*FR*HW)        // [8][7][4096]  c_match
#define WS_PART  (WS_CM  + BB*FR*HW)        // [8*7][512]    gs partial sums
#define WS_VSUM  (WS_PART + BB*FR*NPART)    // [56]
#define WS_GS    (WS_VSUM + BB*FR)          // [56]

// ---------------- kernel A: zero the v_sum accumulators ----------------
__global__ void k_init(float* ws) {
    int t = threadIdx.x;
    if (t < BB*FR) ws[WS_VSUM + t] = 0.0f;
}

// ------- kernel B: 4x bilinear downsample + threshold masks, v_sum -------
// v_t_r[b,p], v_r[b,f,p]; v_sum[b,f] += v_t_r*v_r (exact integer counts)
__global__ void k_masks(const float* __restrict__ v_t,
                        const float* __restrict__ v_al,
                        float* __restrict__ ws) {
    int b     = blockIdx.x >> 4;           // 8 b * 16 chunks
    int chunk = blockIdx.x & 15;
    int tid   = threadIdx.x;
    int p     = chunk * 256 + tid;
    int i = p >> 6, j = p & 63;
    int r = 4*i + 1, c = 4*j + 1;

    const float* vt = v_t + (size_t)b * HIN * HIN;
    float a = 0.25f * (vt[r*HIN + c] + vt[r*HIN + c + 1] +
                       vt[(r+1)*HIN + c] + vt[(r+1)*HIN + c + 1]);
    float vtr = (a > 0.5f) ? 1.0f : 0.0f;
    ws[WS_VTR + b*HW + p] = vtr;

    int lane = tid & 31;
    #pragma unroll
    for (int f = 0; f < FR; ++f) {
        const float* va = v_al + (size_t)(b*FR + f) * HIN * HIN;
        float m = 0.25f * (va[r*HIN + c] + va[r*HIN + c + 1] +
                           va[(r+1)*HIN + c] + va[(r+1)*HIN + c + 1]);
        float vr = (m > 0.5f) ? 1.0f : 0.0f;
        ws[WS_VR + (b*FR + f)*HW + p] = vr;
        float s = vtr * vr;                       // 0/1 -> exact int sums
        s += __shfl_xor(s, 16, 32);
        s += __shfl_xor(s, 8, 32);
        s += __shfl_xor(s, 4, 32);
        s += __shfl_xor(s, 2, 32);
        s += __shfl_xor(s, 1, 32);
        if (lane == 0) atomicAdd(&ws[WS_VSUM + b*FR + f], s);
    }
}

// ------- kernel C: gs partials via V_WMMA_F32_16X16X4_F32 reduction -------
// One wave = (b, c-half of 64 ch, 16-pixel tile). A[m,k] = vmap*c0*c_rest for
// pixel m, channel k; B = ones -> D rows accumulate per-pixel channel sums on
// the matrix pipe. 512 blocks x 256 threads = 4096 waves, c_feats read ONCE.
__global__ void k_gs_wmma(const float* __restrict__ cf,
                          float* __restrict__ ws) {
    int wib    = threadIdx.x >> 5;                       // wave in block
    int lane   = threadIdx.x & 31;
    int waveId = blockIdx.x * 8 + wib;                   // 0..4095
    int b      = waveId >> 9;
    int rem    = waveId & 511;
    int chalf  = rem >> 8;                               // 0/1
    int ptile  = rem & 255;
    int cstart = chalf * 64;
    int pix    = ptile * 16 + (lane & 15);
    int choff  = (lane >> 4) << 1;                       // K pair: 0 or 2

    // vmap per f for this lane's pixel (reused across whole c loop)
    float vtr = ws[WS_VTR + b*HW + pix];
    float vmapv[FR];
    #pragma unroll
    for (int f = 0; f < FR; ++f)
        vmapv[f] = vtr * ws[WS_VR + (b*FR + f)*HW + pix];

    v2f bones = {1.0f, 1.0f};
    v8f acc[FR];
    #pragma unroll
    for (int f = 0; f < FR; ++f) acc[f] = (v8f){0.f,0.f,0.f,0.f,0.f,0.f,0.f,0.f};

    for (int c = cstart; c < cstart + 64; c += 4) {      // wave-uniform, EXEC full
        int ch = c + choff;
        long base = ((long)(b*CC + ch) * FF) * HW + pix; // c_feats[b][ch][0][pix]
        float c0a = cf[base];
        float c0b = cf[base + (long)FF*HW];
        #pragma unroll
        for (int f = 0; f < FR; ++f) {
            float ra = cf[base + (long)(1 + f)*HW];
            float rb = cf[base + (long)FF*HW + (long)(1 + f)*HW];
            v2f a;
            a.x = vmapv[f] * c0a * ra;
            a.y = vmapv[f] * c0b * rb;
            acc[f] = __builtin_amdgcn_wmma_f32_16x16x4_f32(
                false, a, false, bones, (short)0, acc[f], false, false);
        }
    }

    // D columns are identical row sums; fold 8 VGPRs + lo/hi halves.
    #pragma unroll
    for (int f = 0; f < FR; ++f) {
        float s = acc[f][0] + acc[f][1] + acc[f][2] + acc[f][3] +
                  acc[f][4] + acc[f][5] + acc[f][6] + acc[f][7];
        s += __shfl_xor(s, 16, 32);                      // M 0-7 + M 8-15
        if (lane == 0)
            ws[WS_PART + (b*FR + f)*NPART + chalf*256 + ptile] = s;
    }
}

// ------- kernel D: deterministic tree-reduce partials, finalize gs -------
__global__ void k_gs_final(float* __restrict__ ws) {
    __shared__ float sdata[256];
    int bf  = blockIdx.x;                                // 0..55
    int tid = threadIdx.x;
    const float* part = ws + WS_PART + bf*NPART;
    sdata[tid] = part[tid] + part[tid + 256];
    __syncthreads();
    for (int s = 128; s > 0; s >>= 1) {
        if (tid < s) sdata[tid] += sdata[tid + s];
        __syncthreads();
    }
    if (tid == 0) {
        float vsum = ws[WS_VSUM + bf];
        int zero = vsum < 1e-4f;
        float denom = (vsum + (zero ? 1.0f : 0.0f)) * (float)CC;
        ws[WS_GS + bf] = zero ? 0.0f : (sdata[0] / denom);
    }
}

// ------- kernel E: masked softmax over f -> c_match, c_mask channel -------
__global__ void k_softmax(float* __restrict__ ws, float* __restrict__ out) {
    int g   = blockIdx.x * blockDim.x + threadIdx.x;     // 8*4096 threads
    int b   = g >> 12;
    int pix = g & 4095;

    float vr[FR], mv[FR];
    float mx = -1e30f;
    #pragma unroll
    for (int f = 0; f < FR; ++f) {
        vr[f] = ws[WS_VR + (b*FR + f)*HW + pix];
        mv[f] = ws[WS_GS + b*FR + f] * vr[f];
        mx = fmaxf(mx, mv[f]);
    }
    float sum = 0.0f, ex[FR];
    #pragma unroll
    for (int f = 0; f < FR; ++f) { ex[f] = __builtin_expf(mv[f] - mx) * vr[f]; sum += ex[f]; }
    sum += (sum < 1e-4f) ? 1.0f : 0.0f;
    float inv = 1.0f / sum, cmaskacc = 0.0f;
    #pragma unroll
    for (int f = 0; f < FR; ++f) {
        float cm = ex[f] * inv;
        ws[WS_CM + (b*FR + f)*HW + pix] = cm;
        cmaskacc += cm * vr[f];
    }
    float cmask = 1.0f - cmaskacc;
    out[((size_t)b*257 + 256)*HW + pix] = cmask;         // last channel of out0
    out[(size_t)BB*257*HW + (size_t)b*HW + pix] = cmask; // out1
}

// ------- kernel F: c0 passthrough + c_out = sum_f c_rest*c_match -------
// c_feats read once; c_match held in registers across the c loop.
__global__ void k_cout(const float* __restrict__ cf,
                       const float* __restrict__ ws,
                       float* __restrict__ out) {
    int blk    = blockIdx.x;                 // 8 b * 16 pchunk * 4 cquarter
    int b      = blk >> 6;
    int pchunk = (blk >> 2) & 15;
    int q      = blk & 3;
    int p      = pchunk * 256 + threadIdx.x;

    float cm[FR];
    #pragma unroll
    for (int f = 0; f < FR; ++f) cm[f] = ws[WS_CM + (b*FR + f)*HW + p];

    for (int c = q*32; c < q*32 + 32; ++c) {
        long base = ((long)(b*CC + c) * FF) * HW + p;
        float c0v = cf[base];
        float acc = 0.0f;
        #pragma unroll
        for (int f = 0; f < FR; ++f) acc += cf[base + (long)(1 + f)*HW] * cm[f];
        out[((size_t)b*257 + c)*HW + p]       = c0v;
        out[((size_t)b*257 + 128 + c)*HW + p] = acc;
    }
}

extern "C" void kernel_launch(void* const* d_in, const int* in_sizes, int n_in,
                              void* d_out, int out_size, void* d_ws, size_t ws_size,
                              hipStream_t stream) {
    const float* c_feats   = (const float*)d_in[0];
    const float* v_t       = (const float*)d_in[1];
    const float* v_aligned = (const float*)d_in[2];
    float* out = (float*)d_out;
    float* ws  = (float*)d_ws;

    k_init    <<<1,   128, 0, stream>>>(ws);
    k_masks   <<<128, 256, 0, stream>>>(v_t, v_aligned, ws);
    k_gs_wmma <<<512, 256, 0, stream>>>(c_feats, ws);
    k_gs_final<<<56,  256, 0, stream>>>(ws);
    k_softmax <<<128, 256, 0, stream>>>(ws, out);
    k_cout    <<<512, 256, 0, stream>>>(c_feats, ws, out);
}